// Model_25907242730053
// MI455X (gfx1250) — compile-verified
//
#include <hip/hip_runtime.h>
#include <cstdint>

// Problem constants (from reference): in (4,64,224,224) f32, k=3, pad=1, stride=1,
// reference's nh=k+d convention -> out spatial 223x223, out shape (4, 576, 49729).
#define BATCH 4
#define CH    64
#define IH    224
#define IW    224
#define OH    223
#define OW    223
#define KPOS  3              // kernel extent per axis
#define ROWS_PER_BLK 8
#define LDS_ROW_F 228        // floats per LDS row: [0..2] unused, [3]=left zero pad,
                             // [4..227]=x row (16B-aligned interior), stride 912B (16B mult)
#define THREADS 256

__device__ __forceinline__ void async_load_b128_to_lds(uint32_t lds_byte_off,
                                                       const float* gaddr) {
  // per-lane: copies 16 bytes from the global address gaddr to the LDS address lds_byte_off; tracked by ASYNCcnt
  asm volatile("global_load_async_to_lds_b128 %0, %1, off"
               :
               : "v"(lds_byte_off), "v"(gaddr)
               : "memory");
}

__device__ __forceinline__ void async_store_b32_from_lds(float* gaddr,
                                                         uint32_t lds_byte_off) {
  // per-lane: copies 4 bytes from the LDS address lds_byte_off to the global address gaddr; non-temporal (write stream >> L2)
  asm volatile("global_store_async_from_lds_b32 %0, %1, off th:TH_STORE_NT"
               :
               : "v"(gaddr), "v"(lds_byte_off)
               : "memory");
}

__device__ __forceinline__ void wait_asynccnt0() {
  asm volatile("s_wait_asynccnt 0" ::: "memory");
}

__global__ __launch_bounds__(THREADS) void im2col_kernel(
    const float* __restrict__ x, float* __restrict__ out) {
  __shared__ float lds[ROWS_PER_BLK * LDS_ROW_F];

  const int tid = threadIdx.x;

  // Decompose flat block index -> (b, c, i, htile)
  int blk = (int)blockIdx.x;
  const int n_htiles = (OH + ROWS_PER_BLK - 1) / ROWS_PER_BLK;  // 28
  const int htile = blk % n_htiles; blk /= n_htiles;
  const int i     = blk % KPOS;     blk /= KPOS;
  const int c     = blk % CH;       blk /= CH;
  const int b     = blk;
  const int h0    = htile * ROWS_PER_BLK;
  const int nrows = (OH - h0 < ROWS_PER_BLK) ? (OH - h0) : ROWS_PER_BLK;

  // --- 1) zero LDS (covers left-pad column and out-of-range input rows) ---
  {
    float4* lds4 = (float4*)lds;
    const int n4 = (ROWS_PER_BLK * LDS_ROW_F) / 4;  // 456
    for (int t = tid; t < n4; t += THREADS) lds4[t] = make_float4(0.f, 0.f, 0.f, 0.f);
  }
  __syncthreads();

  // --- 2) async-stage input rows: LDS row `row` holds x[b,c, h0+row+i-1, :] at [4..227] ---
  const float* xbase = x + (size_t)(b * CH + c) * IH * IW;
  const uint32_t lds_base = (uint32_t)(uintptr_t)(void*)&lds[0];
  {
    const int chunks_per_row = IW / 4;               // 56 x b128
    const int chunks = nrows * chunks_per_row;
    for (int t = tid; t < chunks; t += THREADS) {
      const int row = t / chunks_per_row;
      const int k   = t - row * chunks_per_row;
      const int r   = h0 + row + i - 1;              // input row index
      if (r >= 0 && r < IH) {
        const uint32_t loff = lds_base + (uint32_t)(row * LDS_ROW_F + 4 + 4 * k) * 4u;
        const float* g = xbase + (size_t)r * IW + 4 * k;   // 16B aligned
        async_load_b128_to_lds(loff, g);
      }
    }
  }
  wait_asynccnt0();     // own wave's async loads landed in LDS
  __syncthreads();      // all waves' loads visible

  // --- 3) stream output rows from LDS: out[b, c*9+i*3+j, (h0+row)*223 + w] = lds[row][3+j+w] ---
  const size_t out_k_base = (size_t)b * (CH * KPOS * KPOS) + (size_t)c * 9 + (size_t)i * 3;
  const int per_row = KPOS * OW;                     // 669
  const int total   = nrows * per_row;
  for (int t = tid; t < total; t += THREADS) {
    const int row = t / per_row;
    const int rem = t - row * per_row;
    const int j   = rem / OW;
    const int w   = rem - j * OW;
    const uint32_t loff = lds_base + (uint32_t)(row * LDS_ROW_F + 3 + j + w) * 4u;
    float* g = out + (out_k_base + (size_t)j) * (size_t)(OH * OW)
                   + (size_t)(h0 + row) * OW + (size_t)w;
    async_store_b32_from_lds(g, loff);
  }
  wait_asynccnt0();     // drain async stores before wave exit
}

extern "C" void kernel_launch(void* const* d_in, const int* in_sizes, int n_in,
                              void* d_out, int out_size, void* d_ws, size_t ws_size,
                              hipStream_t stream) {
  (void)in_sizes; (void)n_in; (void)d_ws; (void)ws_size; (void)out_size;
  const float* x = (const float*)d_in[0];
  float* out = (float*)d_out;

  const int n_htiles = (OH + ROWS_PER_BLK - 1) / ROWS_PER_BLK;         // 28
  const int nblocks  = BATCH * CH * KPOS * n_htiles;                   // 21504
  im2col_kernel<<<dim3(nblocks), dim3(THREADS), 0, stream>>>(x, out);
}